// Model_11888469475838
// MI455X (gfx1250) — compile-verified
//
#include <hip/hip_runtime.h>

// Fused: circular-pad im2col -> h=relu(W1@patches+b1) -> y=W2@h+b2 -> fold(overlap-add) -> crop
// GEMMs use V_WMMA_F32_16X16X4_F32 (fp32, wave32). Interior x-tiles staged to LDS by the
// Tensor Data Mover (tensor_load_to_lds + s_wait_tensorcnt); edge tiles use a scalar path.

typedef float v2f  __attribute__((ext_vector_type(2)));
typedef float v8f  __attribute__((ext_vector_type(8)));
typedef unsigned int u32x4 __attribute__((ext_vector_type(4)));
typedef int   i32x4 __attribute__((ext_vector_type(4)));
typedef int   i32x8 __attribute__((ext_vector_type(8)));

constexpr int Bsz = 8, C = 128, L = 16384;
constexpr int KW = 7, STRIDE = 3, PADW = 3;
constexpr int IC = 64, CK = C * KW;                // 896
constexpr int LP = L + 2 * PADW;                   // 16390
constexpr int P = (LP - KW) / STRIDE + 1;          // 5462
constexpr int PT = 32;                             // positions per workgroup tile
constexpr int TILES = (P + PT - 1) / PT;           // 171
constexpr int JSPAN = STRIDE * PT + (KW - STRIDE); // 100 xp-elements per channel per tile
constexpr int XS_STRIDE = JSPAN;                   // contiguous rows (TDM-compatible)
constexpr int HS_STRIDE = 36;                      // h row stride
constexpr int NTHREADS = 256;                      // 8 wave32

#if defined(__has_builtin)
#if __has_builtin(__builtin_amdgcn_tensor_load_to_lds)
#define HAVE_TDM 1
#endif
#endif
#ifndef HAVE_TDM
#warning "CDNA5 probe: __builtin_amdgcn_tensor_load_to_lds NOT available; scalar staging only"
#endif

__global__ __launch_bounds__(NTHREADS)
void fused_conv_mlp_fold(const float* __restrict__ x,
                         const float* __restrict__ W1,
                         const float* __restrict__ b1,
                         const float* __restrict__ W2,
                         const float* __restrict__ b2,
                         float* __restrict__ out)
{
    __shared__ float xs[C * XS_STRIDE];   // staged x tile; later reused as fold accumulator
    __shared__ float hs[IC * HS_STRIDE];  // hidden activations h[64][PT]

    const int tile = blockIdx.x;
    const int b    = blockIdx.y;
    const int p0   = tile * PT;
    const int pcnt = min(PT, P - p0);

    const int tid  = threadIdx.x;
    const int wave = tid >> 5;
    const int lane = tid & 31;
    const int half = lane >> 4;   // lane group 0-15 / 16-31
    const int l16  = lane & 15;

    // ---- stage x tile into LDS ----
    const int gstart = 3 * p0 - PADW;                       // first x index needed
    const bool interior = (gstart >= 0) && (gstart + JSPAN <= L);
#if defined(HAVE_TDM)
    if (interior) {
        if (wave == 0) {
            const unsigned int lds_off = (unsigned int)(uintptr_t)(void*)xs;
            const unsigned long long ga =
                (unsigned long long)(uintptr_t)(x + ((size_t)b * C) * L + gstart);
            u32x4 g0;
            g0[0] = 1u;                                     // count=1, flags=0
            g0[1] = lds_off;                                // lds_addr
            g0[2] = (unsigned int)ga;                       // global_addr[31:0]
            g0[3] = ((unsigned int)(ga >> 32) & 0x1FFFFFFu) // global_addr[56:32]
                    | (2u << 30);                           // type=2 ("image")
            i32x8 g1;
            const unsigned int td0 = JSPAN, td1 = C;        // tensor dims (no OOB)
            const unsigned long long s0 = (unsigned long long)L; // dim0 stride (elements)
            g1[0] = (int)(2u << 16);                        // data_size=4B, mask=0, no pad
            g1[1] = (int)((td0 & 0xFFFFu) << 16);           // tensor_dim0[15:0]
            g1[2] = (int)(((td0 >> 16) & 0xFFFFu) | ((td1 & 0xFFFFu) << 16));
            g1[3] = (int)(((td1 >> 16) & 0xFFFFu) | ((unsigned)JSPAN << 16)); // tile_dim0
            g1[4] = (int)((unsigned)C & 0xFFFFu);           // tile_dim1=128, tile_dim2=0
            g1[5] = (int)(unsigned int)(s0 & 0xFFFFFFFFull);
            g1[6] = (int)(unsigned int)((s0 >> 32) & 0xFFFFu); // stride1=0
            g1[7] = 0;
            i32x4 z4 = {0, 0, 0, 0};
#if __clang_major__ >= 23
            i32x8 z8 = {0, 0, 0, 0, 0, 0, 0, 0};
            __builtin_amdgcn_tensor_load_to_lds(g0, g1, z4, z4, z8, 0);
#else
            __builtin_amdgcn_tensor_load_to_lds(g0, g1, z4, z4, 0);
#endif
            __builtin_amdgcn_s_wait_tensorcnt(0);
        }
    } else
#endif
    {
        for (int idx = tid; idx < C * JSPAN; idx += NTHREADS) {
            int c = idx / JSPAN;
            int j = idx - c * JSPAN;
            int l = 3 * p0 + j;                 // coordinate in xp
            float v = 0.f;
            if (l < LP) {
                int src = l - PADW;
                if (src < 0) src += L; else if (src >= L) src -= L;
                v = x[((size_t)b * C + c) * L + src];
            }
            xs[c * XS_STRIDE + j] = v;
        }
    }
    __syncthreads();

    // ---- GEMM1: h = relu(W1[64,896] @ patches[896,PT] + b1) ----
    {
        const int mt = wave & 3;          // 4 tiles along M=64
        const int nt = wave >> 2;         // 2 tiles along N=32
        const int pp = nt * 16 + l16;

        // Compile-time B-fragment offset tables for one 4-channel group
        // (28 ck values = 7 WMMA k-steps). t = 4u + 2*half (+1):
        // offset = (t/7)*XS_STRIDE + t%7, added to base &xs[3*pp].
        static constexpr int OFF0_H0[7] = {0, 4, 101, 105, 202, 206, 303};
        static constexpr int OFF1_H0[7] = {1, 5, 102, 106, 203, 300, 304};
        static constexpr int OFF0_H1[7] = {2, 6, 103, 200, 204, 301, 305};
        static constexpr int OFF1_H1[7] = {3, 100, 104, 201, 205, 302, 306};

        const float* xbase = &xs[3 * pp];
        const float* bp0[7];
        const float* bp1[7];
#pragma unroll
        for (int u = 0; u < 7; ++u) {
            bp0[u] = xbase + (half ? OFF0_H1[u] : OFF0_H0[u]);
            bp1[u] = xbase + (half ? OFF1_H1[u] : OFF1_H0[u]);
        }

        const float* ap = W1 + (size_t)(mt * 16 + l16) * CK + 2 * half;
        v8f acc = {};
#pragma unroll 8
        for (int g = 0; g < C / 4; ++g) {          // 32 groups x 7 k-steps = 224 WMMAs
#pragma unroll
            for (int u = 0; u < 7; ++u) {
                v2f a = *(const v2f*)(ap + g * 28 + 4 * u);  // A[l16][2*half + r]
                v2f bv;                                      // B[2*half+r][l16]
                bv[0] = bp0[u][g * (4 * XS_STRIDE)];
                bv[1] = bp1[u][g * (4 * XS_STRIDE)];
                acc = __builtin_amdgcn_wmma_f32_16x16x4_f32(false, a, false, bv,
                                                            (short)0, acc, false, false);
            }
        }
        const int mbase = mt * 16 + 8 * half;        // D[r + 8*half][l16]
#pragma unroll
        for (int r = 0; r < 8; ++r) {
            float v = acc[r] + b1[mbase + r];
            hs[(mbase + r) * HS_STRIDE + pp] = fmaxf(v, 0.f);
        }
    }
    __syncthreads();

    // ---- reuse xs as fold accumulator ----
    for (int idx = tid; idx < C * XS_STRIDE; idx += NTHREADS) xs[idx] = 0.f;
    __syncthreads();

    // ---- GEMM2: y = W2[896,64] @ h[64,PT] + b2 ; fold into xs via LDS atomics ----
    for (int id = wave; id < (CK / 16) * 2; id += 8) {        // 112 output tiles
        const int mt = id >> 1;
        const int nt = id & 1;
        const int pp = nt * 16 + l16;
        const float* arow = W2 + (size_t)(mt * 16 + l16) * IC + 2 * half;
        const float* hcol = &hs[2 * half * HS_STRIDE + pp];
        v8f acc = {};
#pragma unroll
        for (int kk = 0; kk < IC; kk += 4) {
            v2f a = *(const v2f*)(arow + kk);
            v2f bv;
            bv[0] = hcol[kk * HS_STRIDE];
            bv[1] = hcol[(kk + 1) * HS_STRIDE];
            acc = __builtin_amdgcn_wmma_f32_16x16x4_f32(false, a, false, bv,
                                                        (short)0, acc, false, false);
        }
        if (pp < pcnt) {                                      // mask tail positions
            const int mbase = mt * 16 + 8 * half;
#pragma unroll
            for (int r = 0; r < 8; ++r) {
                const int ck = mbase + r;                     // y row index = c*7 + k
                const float v = acc[r] + b2[ck];
                const int cc = ck / KW, kk2 = ck - cc * KW;
                atomicAdd(&xs[cc * XS_STRIDE + 3 * pp + kk2], v);  // ds_add_f32
            }
        }
    }
    __syncthreads();

    // ---- write folded tile; 4-element tile-boundary overlap uses global atomics ----
    for (int idx = tid; idx < C * JSPAN; idx += NTHREADS) {
        int c = idx / JSPAN;
        int j = idx - c * JSPAN;
        int o = 3 * p0 + j - PADW;                            // output index (crop)
        if (o < 0 || o >= L) continue;
        float v = xs[c * XS_STRIDE + j];
        float* dst = out + ((size_t)b * C + c) * L + o;
        if (j < (KW - STRIDE) || j >= STRIDE * PT) atomicAdd(dst, v);
        else *dst = v;
    }
}

extern "C" void kernel_launch(void* const* d_in, const int* in_sizes, int n_in,
                              void* d_out, int out_size, void* d_ws, size_t ws_size,
                              hipStream_t stream) {
    const float* x  = (const float*)d_in[0];
    const float* W1 = (const float*)d_in[1];
    const float* b1 = (const float*)d_in[2];
    const float* W2 = (const float*)d_in[3];
    const float* b2 = (const float*)d_in[4];
    float* out = (float*)d_out;

    // fold is additive onto a zeroed output (harness poisons d_out)
    hipMemsetAsync(out, 0, (size_t)out_size * sizeof(float), stream);

    dim3 grid(TILES, Bsz);
    fused_conv_mlp_fold<<<grid, NTHREADS, 0, stream>>>(x, W1, b1, W2, b2, out);
}